// kernel_generated_1_62904091017464
// MI455X (gfx1250) — compile-verified
//
#include <hip/hip_runtime.h>

typedef float v2f __attribute__((ext_vector_type(2)));
typedef float v8f __attribute__((ext_vector_type(8)));

#define NB   128
#define CIN  256
#define HH   28
#define WW   28
#define PB   4                      // h-rows per workgroup -> M = 112 = 7*16
#define LDS_ROW 132                 // 128 channels + 4 pad (bank-conflict avoidance)
#define T_IN_WORDS  (PB*WW*LDS_ROW) // 14784 floats
#define OUT_S_WORDS (CIN*PB*WW)     // 28672 floats

__global__ __launch_bounds__(256)
void fused_groupconv_wmma(const float* __restrict__ x,
                          const float* __restrict__ Wt,
                          float* __restrict__ out) {
  extern __shared__ float smem[];
  float* tIn  = smem;               // [p][w][ch] with row stride LDS_ROW
  float* outS = smem + T_IN_WORDS;  // [c=256][p=4][w=28]

  const int tid   = threadIdx.x;
  const int wave  = tid >> 5;
  const int lane  = tid & 31;
  const int lhalf = lane & 15;
  const int msel  = (lane >> 4) << 1;    // 0 (lanes 0-15) or 2 (lanes 16-31)
  const int n     = blockIdx.y;
  const int p0    = blockIdx.x * PB;

  // ---- Phase 1: global -> LDS, folding the 2-way channel sum ----
  for (int idx = tid; idx < 128*PB*WW; idx += 256) {
    int ch  = idx / (PB*WW);
    int rem = idx - ch*(PB*WW);
    int p   = rem / WW;
    int w   = rem - p*WW;
    const float* g = x + (((size_t)n*CIN + ch)*HH + (p0 + p))*WW + w;
    tIn[(p*WW + w)*LDS_ROW + ch] = g[0] + g[128*HH*WW];
  }
  __syncthreads();

  // per-lane (p,o) for the 7 M-tiles of the 112-row GEMM
  int pl[7], ow[7];
#pragma unroll
  for (int t = 0; t < 7; ++t) {
    int row = t*16 + lhalf;
    pl[t] = row / WW;
    ow[t] = row - pl[t]*WW;
  }

  // ---- Phase 2: per-wave f32 WMMA GEMMs, 4 groups per wave ----
  for (int jj = 0; jj < 4; ++jj) {
    int j = wave*4 + jj;

    // B fragments for all 9 (i,l) K-steps: K=m on lanes (msel picks m pair), N=k
    v2f bf[9];
    bool kval = (lhalf < 8);
#pragma unroll
    for (int i = 0; i < 3; ++i) {
#pragma unroll
      for (int l = 0; l < 3; ++l) {
        v2f b; b.x = 0.f; b.y = 0.f;
        if (kval) {
          const float* wb = Wt + (((j*3 + i)*4 + msel)*3 + l)*8 + lhalf;
          b.x = wb[0];
          b.y = wb[3*8];   // next m
        }
        bf[i*3 + l] = b;
      }
    }

    v8f acc[7];
#pragma unroll
    for (int t = 0; t < 7; ++t) { v8f z = {}; acc[t] = z; }

    for (int s = 0; s < 9; ++s) {
      int i  = s / 3;
      int l  = s - i*3;
      int cb = j + i - 1;                  // group-axis zero pad: skip step
      if (cb < 0 || cb >= 32) continue;
      int chOff = cb*4 + msel;
#pragma unroll
      for (int t = 0; t < 7; ++t) {
        int pos = ow[t] + l - 1;           // w-tap validity (post-roll pad)
        v2f a; a.x = 0.f; a.y = 0.f;
        if ((unsigned)pos < 28u) {
          int wi = (pos == 0) ? 27 : pos - 1;   // (pos-1) mod 28 from the roll
          a = *(const v2f*)&tIn[(pl[t]*WW + wi)*LDS_ROW + chOff];
        }
        acc[t] = __builtin_amdgcn_wmma_f32_16x16x4_f32(
            false, a, false, bf[s], (short)0, acc[t], false, false);
      }
    }

    // stage D tiles (M=row, N=k) into LDS out buffer
    if (lhalf < 8) {
      int c = j*8 + lhalf;
#pragma unroll
      for (int t = 0; t < 7; ++t) {
#pragma unroll
        for (int r = 0; r < 8; ++r) {
          int rr  = (lane < 16) ? r : (8 + r);
          int row = t*16 + rr;
          int p   = row / WW;
          int o   = row - p*WW;
          outS[c*(PB*WW) + p*WW + o] = acc[t][r];
        }
      }
    }
  }
  __syncthreads();

  // ---- Phase 3: w-coalesced stores with the output h-roll ----
  for (int r = wave; r < CIN*PB; r += 8) {
    int c = r >> 2;
    int p = r & 3;
    int h = p0 + p + 1; if (h >= 28) h -= 28;
    if (lane < WW) {
      out[(((size_t)n*CIN + c)*HH + h)*WW + lane] = outS[c*(PB*WW) + p*WW + lane];
    }
  }
}

extern "C" void kernel_launch(void* const* d_in, const int* in_sizes, int n_in,
                              void* d_out, int out_size, void* d_ws, size_t ws_size,
                              hipStream_t stream) {
  const float* x  = (const float*)d_in[0];   // (128,256,28,28) f32
  const float* Wt = (const float*)d_in[1];   // (32,3,4,3,8)   f32
  float* out = (float*)d_out;                // (128,256,28,28) f32
  (void)in_sizes; (void)n_in; (void)out_size; (void)d_ws; (void)ws_size;

  size_t shmem = (size_t)(T_IN_WORDS + OUT_S_WORDS) * sizeof(float); // 173,824 B
  fused_groupconv_wmma<<<dim3(HH/PB, NB, 1), 256, shmem, stream>>>(x, Wt, out);
}